// GoogleMoleculeNet_75814762709731
// MI455X (gfx1250) — compile-verified
//
#include <hip/hip_runtime.h>
#include <hip/hip_bf16.h>
#include <math.h>

// ---------------------------------------------------------------------------
// GCN (max-aggr) pipeline for MI455X / gfx1250 (wave32, WMMA).
//   - dense transforms: V_WMMA_F32_16X16X4_F32, B operand staged in LDS,
//     K/NCOLS compile-time -> fully unrolled, branch-free inner loop
//   - scatter-max: native global_atomic_max_i32 / min_u32 (sign trick)
//   - pooling/degree: global_atomic_add_f32
// ---------------------------------------------------------------------------

typedef float v2f __attribute__((ext_vector_type(2)));
typedef float v8f __attribute__((ext_vector_type(8)));

__device__ __forceinline__ void atomicMaxF32(float* addr, float val) {
    // IEEE-order-preserving integer trick; dst initialized to -inf.
    if (val >= 0.0f) atomicMax((int*)addr, __float_as_int(val));
    else             atomicMin((unsigned int*)addr, (unsigned int)__float_as_int(val));
}

__global__ void k_fill(float* __restrict__ p, long long n, float v) {
    long long i = blockIdx.x * (long long)blockDim.x + threadIdx.x;
    if (i < n) p[i] = v;
}

// fill only the live `feats` columns of an [n x ld] padded buffer
__global__ void k_fill_cols(float* __restrict__ p, int ld, int feats,
                            long long n, float v) {
    long long t = blockIdx.x * (long long)blockDim.x + threadIdx.x;
    if (t >= n * feats) return;
    long long node = t / feats;
    int f = (int)(t - node * feats);
    p[node * ld + f] = v;
}

__global__ void k_deg(const long long* __restrict__ dst, float* __restrict__ deg, long long E) {
    long long i = blockIdx.x * (long long)blockDim.x + threadIdx.x;
    if (i < E) atomicAdd(&deg[dst[i]], 1.0f);
}

__global__ void k_rsqrt(float* __restrict__ p, long long n) {
    long long i = blockIdx.x * (long long)blockDim.x + threadIdx.x;
    if (i < n) p[i] = rsqrtf(p[i]);
}

// Out[rows x NCOLS] = A[rows x K] @ W[K x NCOLS] (+bias+relu if ACT==1)
// One wave per 16x16 tile, 8 waves/block. Requires rows % 16 == 0 (true for
// all call sites: N = 1M, G = 50000). W col-tile staged zero-padded in LDS.
// A 16x4 layout : lane(0-15) K={0,1}, lane(16-31) K={2,3}, row = lane&15
// B 4x16 layout : VGPR0 = rows {0,2}, VGPR1 = rows {1,3}, col = lane&15
// D layout      : VGPR v -> row v + 8*(lane>>4), col = lane&15
template <int K, int NCOLS, int ACT>
__global__ void k_wmma_gemm(const float* __restrict__ A, int lda,
                            const float* __restrict__ W,
                            float* __restrict__ Out, int ldo,
                            const float* __restrict__ bias, int rows)
{
    constexpr int KP = (K + 3) & ~3;          // K padded to multiple of 4
    __shared__ float sW[KP * 16];

    const int colBase = blockIdx.y * 16;
    // cooperative stage of W[:, colBase:colBase+16], zero-padded both dims
    for (int t = threadIdx.x; t < KP * 16; t += blockDim.x) {
        const int k = t >> 4, c = t & 15;
        float v = 0.0f;
        if (k < K && (colBase + c) < NCOLS) v = W[k * NCOLS + colBase + c];
        sW[t] = v;
    }
    __syncthreads();                          // all waves participate, then:

    const int wave = threadIdx.x >> 5;
    const int lane = threadIdx.x & 31;
    const int rowBase = (blockIdx.x * (blockDim.x >> 5) + wave) * 16;
    if (rowBase >= rows) return;              // wave-uniform; EXEC stays full

    const int half = lane >> 4;               // 0 or 1
    const int m    = lane & 15;
    const int kh   = half * 2;                // per-lane K sub-offset
    const float* __restrict__ Arow = A + (long long)(rowBase + m) * lda;
    // branch-free tail mask (only used when K % 4 != 0)
    const float tm0 = (KP - 4 + kh     < K) ? 1.0f : 0.0f;
    const float tm1 = (KP - 4 + kh + 1 < K) ? 1.0f : 0.0f;

    v8f acc = {};
#pragma unroll
    for (int k = 0; k < KP; k += 4) {
        const int k0 = k + kh;
        v2f a, b;
        if (k + 4 <= K) {                     // compile-time per iteration
            a.x = Arow[k0];
            a.y = Arow[k0 + 1];
        } else {                              // tail: clamp + mask, no branch
            a.x = Arow[k0     < K ? k0     : K - 1] * tm0;
            a.y = Arow[k0 + 1 < K ? k0 + 1 : K - 1] * tm1;
        }
        b.x = sW[k0 * 16 + m];
        b.y = sW[(k0 + 1) * 16 + m];
        acc = __builtin_amdgcn_wmma_f32_16x16x4_f32(
                  /*neg_a=*/false, a, /*neg_b=*/false, b,
                  /*c_mod=*/(short)0, acc, /*reuse_a=*/false, /*reuse_b=*/false);
    }

    const int col = colBase + m;
    if (col < NCOLS) {                        // only last col-tile is partial
        const float bv = (ACT == 1) ? bias[col] : 0.0f;
#pragma unroll
        for (int v = 0; v < 8; ++v) {
            float val = acc[v] + bv;
            if (ACT == 1) val = fmaxf(val, 0.0f);
            Out[(long long)(rowBase + v + 8 * half) * ldo + col] = val;
        }
    }
}

// One wave per edge (incl. N implicit self-loops); lanes cover features.
__global__ void k_scatter_max(const long long* __restrict__ src,
                              const long long* __restrict__ dst,
                              const float* __restrict__ dinv,
                              const float* __restrict__ xw, int ldx,
                              float* __restrict__ out, int ldo,
                              int feats, long long E, long long N)
{
    long long t = blockIdx.x * (long long)blockDim.x + threadIdx.x;
    long long edge = t >> 5;
    int lane = (int)(t & 31);
    if (edge >= E + N) return;
    long long s, d;
    if (edge < E) { s = src[edge]; d = dst[edge]; }
    else          { s = d = edge - E; }                 // self-loop
    const float nrm = dinv[s] * dinv[d];
    const float* __restrict__ xs = xw + s * ldx;
    float* __restrict__ od = out + d * ldo;
    for (int f = lane; f < feats; f += 32)
        atomicMaxF32(&od[f], xs[f] * nrm);
}

__global__ void k_bias_selu(float* __restrict__ h, int ld,
                            const float* __restrict__ b, int feats, int n)
{
    int t = blockIdx.x * blockDim.x + threadIdx.x;
    if (t >= n * feats) return;
    int node = t / feats, f = t - node * feats;
    float x = h[(long long)node * ld + f] + b[f];
    const float kScale = 1.0507009873554805f, kAlpha = 1.6732632423543772f;
    h[(long long)node * ld + f] = x > 0.0f ? kScale * x
                                           : kScale * kAlpha * (expf(x) - 1.0f);
}

__global__ void k_pool(const float* __restrict__ h, int ld,
                       const long long* __restrict__ batch,
                       float* __restrict__ pooled, int feats, int n)
{
    int t = blockIdx.x * blockDim.x + threadIdx.x;
    if (t >= n * feats) return;
    int node = t / feats, f = t - node * feats;
    atomicAdd(&pooled[batch[node] * feats + f], h[(long long)node * ld + f]);
}

static inline long long cdivll(long long a, long long b) { return (a + b - 1) / b; }

extern "C" void kernel_launch(void* const* d_in, const int* in_sizes, int n_in,
                              void* d_out, int out_size, void* d_ws, size_t ws_size,
                              hipStream_t stream)
{
    // ---- inputs (setup_inputs order) ----
    const float*     x     = (const float*)    d_in[0];   // [N,75]
    const long long* eidx  = (const long long*)d_in[1];   // [2,E] int64
    const long long* batch = (const long long*)d_in[2];   // [N]   int64
    const float* W1 = (const float*)d_in[4];  const float* b1 = (const float*)d_in[5];
    const float* W2 = (const float*)d_in[6];  const float* b2 = (const float*)d_in[7];
    const float* W3 = (const float*)d_in[8];  const float* b3 = (const float*)d_in[9];
    const float* W4 = (const float*)d_in[10]; const float* b4 = (const float*)d_in[11];
    const float* lw1 = (const float*)d_in[12]; const float* lb1 = (const float*)d_in[13];
    const float* lw2 = (const float*)d_in[14]; const float* lb2 = (const float*)d_in[15];

    const long long N = in_sizes[0] / 75;
    const long long E = in_sizes[1] / 2;
    const long long G = out_size / 27;
    const long long* src = eidx;
    const long long* dst = eidx + E;

    // ---- workspace carve ----
    const int LD = 48;                         // padded feature stride
    float* deg    = (float*)d_ws;              // [N]  -> becomes dinv
    float* bufA   = deg  + N;                  // [N,48] aggregated activations
    float* bufB   = bufA + N * LD;             // [N,48] xw scratch
    float* pooled = bufB + N * LD;             // [G,36]
    float* z      = pooled + G * 36;           // [G,96]

    const int T = 256;

    // ---- symmetric degree normalization (self-loops => deg >= 1) ----
    k_fill  <<<cdivll(N, T), T, 0, stream>>>(deg, N, 1.0f);
    k_deg   <<<cdivll(E, T), T, 0, stream>>>(dst, deg, E);
    k_rsqrt <<<cdivll(N, T), T, 0, stream>>>(deg, N);

    // ---- 4 GCNConv(max) layers: gemm(A->B), fill(A,-inf), scatter(B->A), selu(A)
    const dim3 blk(256);                               // 8 waves, 128 rows/block
    const unsigned gx = (unsigned)cdivll(N, 128);

    // layer 1: x[75] -> 15
    k_wmma_gemm<75, 15, 0><<<dim3(gx, 1), blk, 0, stream>>>(x, 75, W1, bufB, LD, nullptr, (int)N);
    k_fill_cols<<<cdivll(N * 15, T), T, 0, stream>>>(bufA, LD, 15, N, -INFINITY);
    k_scatter_max<<<cdivll((E + N) * 32, T), T, 0, stream>>>(src, dst, deg, bufB, LD, bufA, LD, 15, E, N);
    k_bias_selu<<<cdivll(N * 15, T), T, 0, stream>>>(bufA, LD, b1, 15, (int)N);

    // layer 2: 15 -> 20
    k_wmma_gemm<15, 20, 0><<<dim3(gx, 2), blk, 0, stream>>>(bufA, LD, W2, bufB, LD, nullptr, (int)N);
    k_fill_cols<<<cdivll(N * 20, T), T, 0, stream>>>(bufA, LD, 20, N, -INFINITY);
    k_scatter_max<<<cdivll((E + N) * 32, T), T, 0, stream>>>(src, dst, deg, bufB, LD, bufA, LD, 20, E, N);
    k_bias_selu<<<cdivll(N * 20, T), T, 0, stream>>>(bufA, LD, b2, 20, (int)N);

    // layer 3: 20 -> 27
    k_wmma_gemm<20, 27, 0><<<dim3(gx, 2), blk, 0, stream>>>(bufA, LD, W3, bufB, LD, nullptr, (int)N);
    k_fill_cols<<<cdivll(N * 27, T), T, 0, stream>>>(bufA, LD, 27, N, -INFINITY);
    k_scatter_max<<<cdivll((E + N) * 32, T), T, 0, stream>>>(src, dst, deg, bufB, LD, bufA, LD, 27, E, N);
    k_bias_selu<<<cdivll(N * 27, T), T, 0, stream>>>(bufA, LD, b3, 27, (int)N);

    // layer 4: 27 -> 36
    k_wmma_gemm<27, 36, 0><<<dim3(gx, 3), blk, 0, stream>>>(bufA, LD, W4, bufB, LD, nullptr, (int)N);
    k_fill_cols<<<cdivll(N * 36, T), T, 0, stream>>>(bufA, LD, 36, N, -INFINITY);
    k_scatter_max<<<cdivll((E + N) * 32, T), T, 0, stream>>>(src, dst, deg, bufB, LD, bufA, LD, 36, E, N);
    k_bias_selu<<<cdivll(N * 36, T), T, 0, stream>>>(bufA, LD, b4, 36, (int)N);

    // ---- global add pool ----
    k_fill<<<cdivll(G * 36, T), T, 0, stream>>>(pooled, G * 36, 0.0f);
    k_pool<<<cdivll(N * 36, T), T, 0, stream>>>(bufA, LD, batch, pooled, 36, (int)N);

    // ---- MLP head (bias+relu fused into WMMA epilogue) ----
    const unsigned gxh = (unsigned)cdivll(G, 128);
    k_wmma_gemm<36, 96, 1><<<dim3(gxh, 6), blk, 0, stream>>>(pooled, 36, lw1, z, 96, lb1, (int)G);
    k_wmma_gemm<96, 27, 1><<<dim3(gxh, 2), blk, 0, stream>>>(z, 96, lw2, (float*)d_out, 27, lb2, (int)G);
}